// ModelClass_8529805050223
// MI455X (gfx1250) — compile-verified
//
#include <hip/hip_runtime.h>

#define GG 256
#define NN 512
#define KNN 6
#define FF 64
#define TT (GG*NN)

typedef __attribute__((ext_vector_type(2))) float v2f;
typedef __attribute__((ext_vector_type(8))) float v8f;

// ---------------------------------------------------------------------------
// Squared norms per node: sq[t] = sum_f x[t][f]^2
// ---------------------------------------------------------------------------
__global__ void sqnorm_kernel(const float* __restrict__ x, float* __restrict__ sq) {
    int t = blockIdx.x * blockDim.x + threadIdx.x;
    if (t >= TT) return;
    const float4* row = (const float4*)(x + (size_t)t * FF);
    float s = 0.f;
#pragma unroll
    for (int i = 0; i < FF/4; ++i) {
        float4 v = row[i];
        s += v.x*v.x + v.y*v.y + v.z*v.z + v.w*v.w;
    }
    sq[t] = s;
}

// ---------------------------------------------------------------------------
// Per-graph kNN via WMMA Gram-matrix tiles.
// One wave per (graph, 16-row strip). For each of 32 column tiles: preload all
// 16 B chunks (pipelined loads), then 16 back-to-back WMMAs; then distance
// correction + register-resident top-6 swap cascade per row.
// ---------------------------------------------------------------------------
__global__ void knn_kernel(const float* __restrict__ x, const float* __restrict__ sq,
                           int* __restrict__ knn_out) {
    int g    = blockIdx.x >> 5;       // 32 row tiles per graph (N=512)
    int rt   = blockIdx.x & 31;
    int base = g * NN;
    int row0 = rt * 16;
    int lane = threadIdx.x;           // blockDim.x == 32 (wave32)
    int half = lane >> 4;
    int lm   = lane & 15;

    __shared__ float dtile[16][17];

    float bestd[KNN];
    int   besti[KNN];
#pragma unroll
    for (int k = 0; k < KNN; ++k) { bestd[k] = __builtin_inff(); besti[k] = 0; }

    float sqr[8];
#pragma unroll
    for (int r = 0; r < 8; ++r) sqr[r] = sq[base + row0 + r + 8*half];

    // A tile (this strip's rows) preloaded once into registers
    const float* xa = x + (size_t)(base + row0 + lm) * FF;
    v2f areg[16];
#pragma unroll
    for (int kt = 0; kt < 16; ++kt) {
        int ka = kt*4 + half*2;
        areg[kt] = v2f{ xa[ka], xa[ka+1] };
    }

    for (int ct = 0; ct < NN/16; ++ct) {
        int col0 = ct * 16;
        const float* xb = x + (size_t)(base + col0 + lm) * FF;

        // phase 1: issue all 16 independent B loads
        v2f breg[16];
#pragma unroll
        for (int kt = 0; kt < 16; ++kt) {
            int ka = kt*4 + half*2;
            breg[kt] = v2f{ xb[ka], xb[ka+1] };
        }
        // phase 2: 16 back-to-back WMMAs
        v8f c = {};
#pragma unroll
        for (int kt = 0; kt < 16; ++kt)
            c = __builtin_amdgcn_wmma_f32_16x16x4_f32(false, areg[kt], false, breg[kt],
                                                      (short)0, c, false, false);

        float sqc = sq[base + col0 + lm];
#pragma unroll
        for (int r = 0; r < 8; ++r) {
            int rl = r + 8*half;
            float d = sqr[r] + sqc - 2.0f * c[r];
            if (row0 + rl == col0 + lm) d = __builtin_inff();  // exclude self
            dtile[rl][lm] = d;
        }
        __syncthreads();  // single-wave WG: DS in-order; barrier is a NOP
        if (lane < 16) {
#pragma unroll
            for (int j = 0; j < 16; ++j) {
                float d  = dtile[lane][j];
                int   ci = base + col0 + j;
#pragma unroll
                for (int k = 0; k < KNN; ++k) {      // sorted insert via swaps
                    if (d < bestd[k]) {
                        float td = bestd[k]; bestd[k] = d;  d  = td;
                        int   ti = besti[k]; besti[k] = ci; ci = ti;
                    }
                }
            }
        }
        __syncthreads();
    }
    if (lane < 16) {
        int t = base + row0 + lane;
#pragma unroll
        for (int k = 0; k < KNN; ++k) knn_out[t*KNN + k] = besti[k];
    }
}

// ---------------------------------------------------------------------------
// Generic fused WMMA GEMM: out = act(A[M,Kd] @ W[Kd,Nd] + bias_scale*bias
//                                    (+ residual))
// act: 0=none, 1=PReLU(per-col a), 2=LeakyReLU(0.2)
// One wave per 16x16 output tile; K processed in 64-wide blocks with a
// preload-all / compute-all split so loads pipeline. Requires Kd % 64 == 0.
// ---------------------------------------------------------------------------
__global__ void wmma_gemm_kernel(const float* __restrict__ A, const float* __restrict__ W,
                                 const float* __restrict__ bias, float bias_scale,
                                 const float* __restrict__ residual,
                                 const float* __restrict__ prelu_a, int act,
                                 float* __restrict__ out, int M, int Kd, int Nd) {
    int lane   = threadIdx.x & 31;
    int wave   = threadIdx.x >> 5;
    int wpb    = blockDim.x >> 5;
    int tilesN = Nd >> 4;
    int tiles  = (M >> 4) * tilesN;
    int gw     = blockIdx.x * wpb + wave;
    int stride = gridDim.x * wpb;
    int half   = lane >> 4;
    int lm     = lane & 15;

    for (int t = gw; t < tiles; t += stride) {
        int rt = t / tilesN, ct = t - rt * tilesN;
        int row0 = rt << 4, col0 = ct << 4;
        const float* arow = A + (size_t)(row0 + lm) * Kd;
        v8f c = {};
        for (int kb0 = 0; kb0 < Kd; kb0 += 64) {
            // phase 1: issue 16 A-chunk + 16 B-chunk loads back-to-back
            v2f areg[16], breg[16];
#pragma unroll
            for (int kt = 0; kt < 16; ++kt) {
                int ka = kb0 + kt*4 + half*2;
                areg[kt] = v2f{ arow[ka], arow[ka+1] };
                breg[kt] = v2f{ W[(size_t)ka*Nd + col0 + lm],
                                W[(size_t)(ka+1)*Nd + col0 + lm] };
            }
            // phase 2: 16 back-to-back WMMAs
#pragma unroll
            for (int kt = 0; kt < 16; ++kt)
                c = __builtin_amdgcn_wmma_f32_16x16x4_f32(false, areg[kt], false, breg[kt],
                                                          (short)0, c, false, false);
        }
        int   col = col0 + lm;
        float bv  = bias ? bias_scale * bias[col] : 0.0f;
        float av  = (act == 1) ? prelu_a[col] : 0.2f;
#pragma unroll
        for (int r = 0; r < 8; ++r) {
            int   row = row0 + r + 8*half;
            float v   = c[r] + bv;
            if (residual) v += residual[(size_t)row*Nd + col];
            if (act == 1)      v = (v >= 0.f) ? v : av * v;
            else if (act == 2) v = (v >= 0.f) ? v : 0.2f * v;
            out[(size_t)row*Nd + col] = v;
        }
    }
}

// ---------------------------------------------------------------------------
// Neighbor gather-sum: s[t][f] = sum_k h[idx[t][k]][f]
// ---------------------------------------------------------------------------
__global__ void gather_kernel(const float* __restrict__ h, const int* __restrict__ idx,
                              float* __restrict__ s) {
    long tid = (long)blockIdx.x * blockDim.x + threadIdx.x;
    if (tid >= (long)TT * FF) return;
    int t = (int)(tid >> 6);
    int f = (int)(tid & 63);
    float acc = 0.f;
#pragma unroll
    for (int k = 0; k < KNN; ++k) {
        int j = idx[t*KNN + k];
        acc += h[(size_t)j*FF + f];
    }
    s[tid] = acc;
}

// ---------------------------------------------------------------------------
// Global add pool: pooled[g][f] = sum_n h[g*N+n][f]   (block = 64 threads)
// ---------------------------------------------------------------------------
__global__ void pool_kernel(const float* __restrict__ h, float* __restrict__ pooled) {
    int g = blockIdx.x;
    int f = threadIdx.x;
    float s = 0.f;
    const float* base = h + (size_t)g * NN * FF + f;
    for (int n = 0; n < NN; ++n) s += base[(size_t)n * FF];
    pooled[g*FF + f] = s;
}

// ---------------------------------------------------------------------------
// Final [G,64] @ [64,1] + b : per-graph dot product reduction
// ---------------------------------------------------------------------------
__global__ void final_kernel(const float* __restrict__ y, const float* __restrict__ w,
                             const float* __restrict__ b, float* __restrict__ out) {
    __shared__ float red[FF];
    int g = blockIdx.x;
    int f = threadIdx.x;
    red[f] = y[g*FF + f] * w[f];
    __syncthreads();
#pragma unroll
    for (int s = FF/2; s > 0; s >>= 1) {
        if (f < s) red[f] += red[f + s];
        __syncthreads();
    }
    if (f == 0) out[g] = red[0] + b[0];
}

// ---------------------------------------------------------------------------
extern "C" void kernel_launch(void* const* d_in, const int* in_sizes, int n_in,
                              void* d_out, int out_size, void* d_ws, size_t ws_size,
                              hipStream_t stream) {
    const float* x   = (const float*)d_in[0];
    const float* W1  = (const float*)d_in[1];  const float* b1  = (const float*)d_in[2];
    const float* a1  = (const float*)d_in[3];
    const float* W2  = (const float*)d_in[4];  const float* b2  = (const float*)d_in[5];
    const float* a2  = (const float*)d_in[6];
    const float* Wc1 = (const float*)d_in[7];  const float* bc1 = (const float*)d_in[8];
    const float* Wc2 = (const float*)d_in[9];  const float* bc2 = (const float*)d_in[10];
    const float* Wc3 = (const float*)d_in[11]; const float* bc3 = (const float*)d_in[12];
    const float* Wc4 = (const float*)d_in[13]; const float* bc4 = (const float*)d_in[14];
    const float* W3  = (const float*)d_in[15]; const float* b3  = (const float*)d_in[16];
    const float* a3  = (const float*)d_in[17];
    const float* W4  = (const float*)d_in[18]; const float* b4  = (const float*)d_in[19];
    const float* a4  = (const float*)d_in[20];
    const float* Wh1 = (const float*)d_in[21]; const float* bh1 = (const float*)d_in[22];
    const float* Wh2 = (const float*)d_in[23]; const float* bh2 = (const float*)d_in[24];
    const float* Wh3 = (const float*)d_in[25]; const float* bh3 = (const float*)d_in[26];
    const float* Wh4 = (const float*)d_in[27]; const float* bh4 = (const float*)d_in[28];

    // workspace carve-up (floats)
    float* ws     = (float*)d_ws;
    float* sq     = ws;                                     // TT
    int*   idx    = (int*)(ws + TT);                        // TT*KNN
    float* bufA   = ws + TT + (size_t)TT*KNN;               // TT*FF
    float* bufB   = bufA + (size_t)TT*FF;                   // TT*FF
    float* bufC   = bufB + (size_t)TT*FF;                   // TT*FF
    float* pooled = bufC + (size_t)TT*FF;                   // GG*FF
    float* y1     = pooled + GG*FF;                         // GG*4FF
    float* y2     = y1 + GG*4*FF;                           // GG*4FF
    float* y3     = y2 + GG*4*FF;                           // GG*FF
    float* outp   = (float*)d_out;

    auto gemm = [&](const float* A, const float* W, const float* bias, float bscale,
                    const float* res, const float* pa, int act, float* out,
                    int M, int Kd, int Nd) {
        int tiles = (M/16) * (Nd/16);
        int grid  = (tiles + 7) / 8;        // 8 waves / 256-thread block
        wmma_gemm_kernel<<<grid, 256, 0, stream>>>(A, W, bias, bscale, res, pa, act,
                                                   out, M, Kd, Nd);
    };

    // kNN graph construction from raw x
    sqnorm_kernel<<<TT/256, 256, 0, stream>>>(x, sq);
    knn_kernel<<<GG*(NN/16), 32, 0, stream>>>(x, sq, idx);

    // encoder MLP
    gemm(x,    W1, b1, 1.f, nullptr, a1, 1, bufA, TT, FF, FF);
    gemm(bufA, W2, b2, 1.f, nullptr, a2, 1, bufB, TT, FF, FF);

    // 4x GeneralConv: out = (sum_nbr h) @ Wc + K*bc + h
    gather_kernel<<<(TT*FF)/256, 256, 0, stream>>>(bufB, idx, bufA);
    gemm(bufA, Wc1, bc1, (float)KNN, bufB, nullptr, 0, bufC, TT, FF, FF);
    gather_kernel<<<(TT*FF)/256, 256, 0, stream>>>(bufC, idx, bufA);
    gemm(bufA, Wc2, bc2, (float)KNN, bufC, nullptr, 0, bufB, TT, FF, FF);
    gather_kernel<<<(TT*FF)/256, 256, 0, stream>>>(bufB, idx, bufA);
    gemm(bufA, Wc3, bc3, (float)KNN, bufB, nullptr, 0, bufC, TT, FF, FF);
    gather_kernel<<<(TT*FF)/256, 256, 0, stream>>>(bufC, idx, bufA);
    gemm(bufA, Wc4, bc4, (float)KNN, bufC, nullptr, 0, bufB, TT, FF, FF);

    // decoder MLP
    gemm(bufB, W3, b3, 1.f, nullptr, a3, 1, bufC, TT, FF, FF);
    gemm(bufC, W4, b4, 1.f, nullptr, a4, 1, bufA, TT, FF, FF);

    // pool + head
    pool_kernel<<<GG, FF, 0, stream>>>(bufA, pooled);
    gemm(pooled, Wh1, bh1, 1.f, nullptr, nullptr, 2, y1, GG, FF,   4*FF);
    gemm(y1,     Wh2, bh2, 1.f, nullptr, nullptr, 2, y2, GG, 4*FF, 4*FF);
    gemm(y2,     Wh3, bh3, 1.f, nullptr, nullptr, 2, y3, GG, 4*FF, FF);
    final_kernel<<<GG, FF, 0, stream>>>(y3, Wh4, bh4, outp);
}